// DSNTDoubleLoss_81252191305977
// MI455X (gfx1250) — compile-verified
//
#include <hip/hip_runtime.h>
#include <math.h>

// Problem constants (from reference setup): B=32, C=8, H=W=256.
#define HW    65536          // elements per heatmap
#define NCH   256            // B*C channels
#define TPB   256            // threads per block (8 wave32s)
#define ITERS 64             // (HW/4)/TPB float4 loads per thread

typedef __attribute__((ext_vector_type(2))) float v2f;
typedef __attribute__((ext_vector_type(8))) float v8f;

// Branchy online-softmax update: the rescale branch (new running max) is rare,
// so the common path is 1 exp + 3 FMAs.
__device__ __forceinline__ void softmax_upd(float x, float wx, float wy,
                                            float& m, float& s, float& sx, float& sy) {
  if (x > m) {
    float c = __expf(m - x);          // exp(-inf)=0 handles the first element
    s  = __fmaf_rn(s,  c, 1.0f);
    sx = __fmaf_rn(sx, c, wx);
    sy = __fmaf_rn(sy, c, wy);
    m  = x;
  } else {
    float e = __expf(x - m);
    s += e;
    sx = __fmaf_rn(e, wx, sx);
    sy = __fmaf_rn(e, wy, sy);
  }
}

__global__ __launch_bounds__(TPB) void dsnt_channel_kernel(
    const float* __restrict__ inp, const float* __restrict__ tgt,
    float* __restrict__ ed_out) {
  const int ch = blockIdx.x;
  const int t  = threadIdx.x;
  const float4* in4 = (const float4*)(inp + (size_t)ch * HW);
  const float4* tg4 = (const float4*)(tgt + (size_t)ch * HW);

  const float inv = 1.0f / 256.0f;
  // For i4 = t + k*256: w = (t&63)*4 + j (constant per thread), h = (t>>6) + 4k.
  const int wbase = (t & 63) * 4;
  const float wx0 = (float)(wbase + 1) * inv;
  const float wx1 = (float)(wbase + 2) * inv;
  const float wx2 = (float)(wbase + 3) * inv;
  const float wx3 = (float)(wbase + 4) * inv;
  const int hrow = t >> 6;

  float m = -INFINITY, s = 0.0f, sx = 0.0f, sy = 0.0f;
  float tmax = -INFINITY;
  int   tidx = 0;

#pragma unroll 4
  for (int k = 0; k < ITERS; ++k) {
    const int i4 = t + (k << 8);
    float4 xv = in4[i4];
    float4 tv = tg4[i4];
    const float wy = (float)(hrow + 4 * k + 1) * inv;
    softmax_upd(xv.x, wx0, wy, m, s, sx, sy);
    softmax_upd(xv.y, wx1, wy, m, s, sx, sy);
    softmax_upd(xv.z, wx2, wy, m, s, sx, sy);
    softmax_upd(xv.w, wx3, wy, m, s, sx, sy);
    const int e0 = i4 << 2;
    if (tv.x > tmax) { tmax = tv.x; tidx = e0;     }  // strict > => first max kept
    if (tv.y > tmax) { tmax = tv.y; tidx = e0 + 1; }
    if (tv.z > tmax) { tmax = tv.z; tidx = e0 + 2; }
    if (tv.w > tmax) { tmax = tv.w; tidx = e0 + 3; }
  }

  // Intra-wave (wave32) butterfly merge of softmax tuple + argmax pair.
#pragma unroll
  for (int off = 16; off > 0; off >>= 1) {
    float mo  = __shfl_xor(m,  off, 32);
    float so  = __shfl_xor(s,  off, 32);
    float sxo = __shfl_xor(sx, off, 32);
    float syo = __shfl_xor(sy, off, 32);
    float M  = fmaxf(m, mo);
    float c1 = __expf(m - M);
    float c2 = __expf(mo - M);
    s  = s  * c1 + so  * c2;
    sx = sx * c1 + sxo * c2;
    sy = sy * c1 + syo * c2;
    m  = M;
    float vo = __shfl_xor(tmax, off, 32);
    int   io = __shfl_xor(tidx, off, 32);
    if (vo > tmax || (vo == tmax && io < tidx)) { tmax = vo; tidx = io; }
  }

  // Cross-wave merge via LDS (8 wave leaders).
  __shared__ float sm[8], ss[8], ssx[8], ssy[8], stv[8];
  __shared__ int   sti[8];
  const int wave = t >> 5;
  if ((t & 31) == 0) {
    sm[wave] = m; ss[wave] = s; ssx[wave] = sx; ssy[wave] = sy;
    stv[wave] = tmax; sti[wave] = tidx;
  }
  __syncthreads();
  if (t == 0) {
    for (int wv = 1; wv < 8; ++wv) {
      float mo = sm[wv], so = ss[wv], sxo = ssx[wv], syo = ssy[wv];
      float M  = fmaxf(m, mo);
      float c1 = __expf(m - M);
      float c2 = __expf(mo - M);
      s  = s  * c1 + so  * c2;
      sx = sx * c1 + sxo * c2;
      sy = sy * c1 + syo * c2;
      m  = M;
      float vo = stv[wv]; int io = sti[wv];
      if (vo > tmax || (vo == tmax && io < tidx)) { tmax = vo; tidx = io; }
    }
    const float px = sx / s;
    const float py = sy / s;
    const float tx = (float)((tidx & 255) + 1) * inv;
    const float ty = (float)((tidx >> 8) + 1) * inv;
    const float dx = tx - px;
    const float dy = ty - py;
    ed_out[ch] = __fsqrt_rn(__fmaf_rn(dx, dx, dy * dy));
  }
}

// Final reduction of the 256 per-channel distances: one wave32, full EXEC.
// Each lane pre-sums 8 values in f32, then one V_WMMA_F32_16X16X4_F32 with
// A = lane values (cols 1,3 zero) and B = ones gives D[m,n] = v[m] + v[m+16];
// a per-lane sum of the 8 D VGPRs + one shfl_xor(16) yields the exact total.
__global__ __launch_bounds__(32) void dsnt_finalize_kernel(
    const float* __restrict__ ed, float* __restrict__ out) {
  const int lane = threadIdx.x;  // 0..31, all lanes active (WMMA needs EXEC=all 1s)
  float p = 0.0f;
#pragma unroll
  for (int k = 0; k < 8; ++k) p += ed[lane + 32 * k];

  v2f a; a.x = p;    a.y = 0.0f;   // A(16x4): col0 = v[0..15], col2 = v[16..31]
  v2f b; b.x = 1.0f; b.y = 1.0f;   // B(4x16) = all ones
  v8f cz = {};
  v8f d = __builtin_amdgcn_wmma_f32_16x16x4_f32(
      /*neg_a=*/false, a, /*neg_b=*/false, b,
      /*c_mod=*/(short)0, cz, /*reuse_a=*/false, /*reuse_b=*/false);

  float sum8 = d[0] + d[1] + d[2] + d[3] + d[4] + d[5] + d[6] + d[7];
  float tot  = sum8 + __shfl_xor(sum8, 16, 32);   // rows 0-7 + rows 8-15
  if (lane == 0) out[0] = tot * (1.0f / 32.0f);   // divide by batch B
}

extern "C" void kernel_launch(void* const* d_in, const int* in_sizes, int n_in,
                              void* d_out, int out_size, void* d_ws, size_t ws_size,
                              hipStream_t stream) {
  const float* inp = (const float*)d_in[0];
  const float* tgt = (const float*)d_in[1];
  float* ws  = (float*)d_ws;   // 256 per-channel distances (fully overwritten each call)
  float* out = (float*)d_out;

  dsnt_channel_kernel<<<NCH, TPB, 0, stream>>>(inp, tgt, ws);
  dsnt_finalize_kernel<<<1, 32, 0, stream>>>(ws, out);
}